// InteractionHead_14293651161363
// MI455X (gfx1250) — compile-verified
//
#include <hip/hip_runtime.h>
#include <stdint.h>

// InteractionHead.preprocess for MI455X (gfx1250, wave32).
// One workgroup per image. Phases:
//   A) issue TDM DMA of the image's 2048 boxes (32KB) into LDS (overlapped)
//   B) build 64-bit sort keys (score bits | idx) and bitonic-sort in LDS
//   C) s_wait_tensorcnt 0, then greedy class-aware NMS in a single wave32:
//      kept list (<=30 entries, caps make more irrelevant) lives in one
//      register set per lane; suppression test = lane-parallel IoU + ballot
//   D) write the 30 fixed output slots (boxes/scores/labels/valid)

#define NB    2048
#define NIMG  32
#define KOUT  30
#define MAXH  15
#define MAXO  15
#define NT    256
#define SCORE_THRESH 0.2f
#define NMS_THRESH   0.5f

typedef unsigned int v4u __attribute__((ext_vector_type(4)));
typedef int          v8i __attribute__((ext_vector_type(8)));
typedef int          v4i __attribute__((ext_vector_type(4)));

#if __has_builtin(__builtin_amdgcn_tensor_load_to_lds) && \
    __has_builtin(__builtin_amdgcn_s_wait_tensorcnt)
#define USE_TDM 1
#endif

__global__ __launch_bounds__(NT)
void hoi_preprocess_kernel(const float* __restrict__ boxes,
                           const float* __restrict__ scores,
                           const int*   __restrict__ labels,
                           float* __restrict__ out)
{
    __shared__ unsigned long long keys[NB];   // 16 KB: (score_bits<<11)|idx
    __shared__ float4             rawbox[NB]; // 32 KB: boxes in original order
    __shared__ short              slab[NB];   //  4 KB: labels in original order
    __shared__ int                slotIdx[KOUT];

    const int tid = threadIdx.x;
    const int img = blockIdx.x;
    const float* gbox = boxes  + (size_t)img * NB * 4;
    const float* gscr = scores + (size_t)img * NB;
    const int*   glab = labels + (size_t)img * NB;

#ifdef USE_TDM
    // ---- Phase A: TDM DMA boxes -> LDS, issued by wave 0, overlapped with sort.
    if (tid < 32) {
        unsigned ldsa = (unsigned)(uintptr_t)(&rawbox[0]);      // LDS byte offset
        unsigned long long ga = (unsigned long long)(uintptr_t)gbox;
        // D# group 0: count=1 | lds_addr | global_addr[56:0] | type=2
        v4u g0;
        g0.x = 1u;
        g0.y = ldsa;
        g0.z = (unsigned)ga;
        g0.w = ((unsigned)((ga >> 32) & 0x01FFFFFFull)) | (2u << 30);
        // D# group 1: data_size=4B, 1-D tensor/tile of 8192 f32 elements
        v8i g1 = {};
        g1[0] = 0x00020000;                 // wg_mask=0, data_size=2 (4 bytes)
        g1[1] = (int)(8192u << 16);         // tensor_dim0[15:0] @ bits 63:48
        g1[2] = (int)(1u << 16);            // tensor_dim0 hi=0; tensor_dim1=1
        g1[3] = (int)(8192u << 16);         // tile_dim0 = 8192 @ bits 127:112
        g1[5] = 8192;                       // tensor_dim0_stride
        v4i gz = {};
#if defined(__clang_major__) && (__clang_major__ >= 23)
        v8i gz8 = {};
        __builtin_amdgcn_tensor_load_to_lds(g0, g1, gz, gz, gz8, 0);
#else
        __builtin_amdgcn_tensor_load_to_lds(g0, g1, gz, gz, 0);
#endif
    }
#else
    for (int i = tid; i < NB; i += NT)
        rawbox[i] = ((const float4*)gbox)[i];
#endif

    // ---- Phase B: build keys + labels, then bitonic sort (descending).
    // Key packing reproduces argsort(where(active,s,-inf))[::-1] exactly:
    // descending by score bits, ties broken by larger original index first,
    // inactive (score < 0.2) forced to the tail.
#if __has_builtin(__builtin_prefetch)
    __builtin_prefetch(glab, 0, 0);
#endif
    for (int i = tid; i < NB; i += NT) {
        float s = gscr[i];
        unsigned sb = (s >= SCORE_THRESH) ? __float_as_uint(s) : 0u;
        keys[i] = (((unsigned long long)sb) << 11) | (unsigned long long)i;
        slab[i] = (short)glab[i];
    }
    if (tid < KOUT) slotIdx[tid] = -1;
    __syncthreads();

    for (int k = 2; k <= NB; k <<= 1) {
        for (int j = k >> 1; j > 0; j >>= 1) {
            for (int i = tid; i < NB; i += NT) {
                int p = i ^ j;
                if (p > i) {
                    unsigned long long a = keys[i], b = keys[p];
                    bool up = (i & k) == 0;          // descending overall
                    if (up ? (a < b) : (a > b)) { keys[i] = b; keys[p] = a; }
                }
            }
            __syncthreads();
        }
    }

#ifdef USE_TDM
    if (tid < 32) __builtin_amdgcn_s_wait_tensorcnt(0);  // DMA done before use
#endif
    __syncthreads();

    // ---- Phase C: greedy class-aware NMS in wave 0.
    // Kept boxes within the 15/15 caps are the only ones that can influence
    // any output slot; each kept entry lives in lane `m`'s registers (m<=29).
    if (tid < 32) {
        const int lane = tid;
        float kx1 = 0.f, ky1 = 0.f, kx2 = 0.f, ky2 = 0.f;
        int   klab = -1;
        int   nk = 0, nh = 0, no = 0;
        for (int i = 0; i < NB; ++i) {
            if (nh >= MAXH && no >= MAXO) break;       // all slots decided
            unsigned long long key = keys[i];
            unsigned sb = (unsigned)(key >> 11);
            if (sb == 0u) break;                       // inactive tail begins
            int idx = (int)(key & 2047ull);
            int l = (int)slab[idx];
            bool isH = (l == 0);
            if (isH ? (nh >= MAXH) : (no >= MAXO)) continue; // cannot affect output
            float4 b = rawbox[idx];
            bool sup = false;
            if (lane < nk && klab == l) {
                float aA = (b.z - b.x) * (b.w - b.y);
                float aB = (kx2 - kx1) * (ky2 - ky1);
                float lx = fmaxf(b.x, kx1), ly = fmaxf(b.y, ky1);
                float rx = fminf(b.z, kx2), ry = fminf(b.w, ky2);
                float iw = fmaxf(rx - lx, 0.0f), ih = fmaxf(ry - ly, 0.0f);
                float inter = iw * ih;
                float iou = inter / (aA + aB - inter + 1e-9f);
                sup = iou > NMS_THRESH;
            }
            if (__ballot(sup) == 0ull) {               // keep candidate
                if (lane == nk) { kx1 = b.x; ky1 = b.y; kx2 = b.z; ky2 = b.w; klab = l; }
                int slot = isH ? nh : (MAXH + no);
                if (lane == 0) slotIdx[slot] = i;
                if (isH) nh++; else no++;
                nk++;
            }
        }
    }
    __syncthreads();

    // ---- Phase D: emit fixed-size outputs (boxes | scores | labels | valid).
    if (tid < KOUT) {
        int i = slotIdx[tid];
        float4 b = make_float4(0.f, 0.f, 0.f, 0.f);
        float sc = 0.f, lv = -1.f, vv = 0.f;
        if (i >= 0) {
            unsigned long long key = keys[i];
            int idx = (int)(key & 2047ull);
            b  = rawbox[idx];
            sc = __uint_as_float((unsigned)(key >> 11));
            lv = (float)slab[idx];
            vv = 1.0f;
        }
        float* ob = out;                              // [32][30][4]
        float* os = out + NIMG * KOUT * 4;            // [32][30]
        float* ol = os  + NIMG * KOUT;                // [32][30]
        float* ov = ol  + NIMG * KOUT;                // [32][30]
        int base = img * KOUT + tid;
        ob[base * 4 + 0] = b.x; ob[base * 4 + 1] = b.y;
        ob[base * 4 + 2] = b.z; ob[base * 4 + 3] = b.w;
        os[base] = sc; ol[base] = lv; ov[base] = vv;
    }
}

extern "C" void kernel_launch(void* const* d_in, const int* in_sizes, int n_in,
                              void* d_out, int out_size, void* d_ws, size_t ws_size,
                              hipStream_t stream) {
    (void)in_sizes; (void)n_in; (void)out_size; (void)d_ws; (void)ws_size;
    const float* boxes  = (const float*)d_in[0];
    const float* scores = (const float*)d_in[1];
    const int*   labels = (const int*)d_in[2];
    float* out = (float*)d_out;
    hipLaunchKernelGGL(hoi_preprocess_kernel, dim3(NIMG), dim3(NT), 0, stream,
                       boxes, scores, labels, out);
}